// FuzzyficationLayer_71038759076543
// MI455X (gfx1250) — compile-verified
//
#include <hip/hip_runtime.h>
#include <hip/hip_bf16.h>

#define BATCH      16384
#define NUM_INPUTS 256
#define NUM_MF     8
#define OUT_COLS   (NUM_INPUTS * NUM_MF)   // 2048
#define BLOCK      256
#define GRID       2048

// -log2(e)/2 : mu = exp(-(d^2)/(2 s^2)) = exp2( d^2 * (-log2(e)/(2 s^2)) )
#define NEG_HALF_LOG2E (-0.7213475204444817f)

typedef float v4f __attribute__((ext_vector_type(4)));
typedef int   v4i __attribute__((vector_size(16)));
// Pointer-to-AS1 / pointer-to-AS3 vector types expected by the async builtin.
typedef __attribute__((address_space(1))) v4i* gvec_p;
typedef __attribute__((address_space(3))) v4i* lvec_p;

#if defined(__HIP_DEVICE_COMPILE__) && __has_builtin(__builtin_amdgcn_global_load_async_to_lds_b128)
#define HAVE_ASYNC_LDS 1
#else
#define HAVE_ASYNC_LDS 0
#endif

__global__ __launch_bounds__(BLOCK)
void fuzzify_kernel(const float* __restrict__ x,
                    const float* __restrict__ centers,
                    const float* __restrict__ sigmas,
                    float* __restrict__ out)
{
    __shared__ float lds_c[OUT_COLS];
    __shared__ float lds_s[OUT_COLS];

    const int tid = threadIdx.x;   // column index j in [0,256)

    // ---- Stage the (tiny) parameter tables into LDS.
#if HAVE_ASYNC_LDS
    // gfx1250 async copy: global -> LDS without touching VGPRs, tracked by ASYNCcnt.
    #pragma unroll
    for (int base = 0; base < OUT_COLS; base += BLOCK * 4) {
        const int o = base + tid * 4;          // 16B per lane
        __builtin_amdgcn_global_load_async_to_lds_b128(
            (gvec_p)(uintptr_t)(const void*)(centers + o),
            (lvec_p)(uintptr_t)(unsigned)(uintptr_t)(void*)(lds_c + o),
            0, 0);
        __builtin_amdgcn_global_load_async_to_lds_b128(
            (gvec_p)(uintptr_t)(const void*)(sigmas + o),
            (lvec_p)(uintptr_t)(unsigned)(uintptr_t)(void*)(lds_s + o),
            0, 0);
    }
    asm volatile("s_wait_asynccnt 0" ::: "memory");
#else
    #pragma unroll
    for (int base = 0; base < OUT_COLS; base += BLOCK * 4) {
        const int o = base + tid * 4;
        *(v4f*)(lds_c + o) = *(const v4f*)(centers + o);
        *(v4f*)(lds_s + o) = *(const v4f*)(sigmas + o);
    }
#endif
    __syncthreads();

    // ---- One-time per-thread register cache of this column's 8 MF params.
    float c[NUM_MF], a2[NUM_MF];
    #pragma unroll
    for (int i = 0; i < NUM_MF; ++i) {
        const float cc = lds_c[tid * NUM_MF + i];
        const float ss = lds_s[tid * NUM_MF + i];
        c[i]  = cc;
        a2[i] = NEG_HALF_LOG2E / (ss * ss);
    }

    // ---- Stream rows: 1 coalesced b32 load, 8 exps, 2 NT b128 stores per thread.
    for (int k = blockIdx.x; k < BATCH; k += GRID) {
        const int kn = k + GRID;
        if (kn < BATCH)
            __builtin_prefetch(&x[(size_t)kn * NUM_INPUTS + tid], 0, 0); // global_prefetch_b8

        const float xv = __builtin_nontemporal_load(&x[(size_t)k * NUM_INPUTS + tid]);

        v4f o0, o1;
        float d;
        d = xv - c[0]; o0.x = __builtin_amdgcn_exp2f(d * d * a2[0]);
        d = xv - c[1]; o0.y = __builtin_amdgcn_exp2f(d * d * a2[1]);
        d = xv - c[2]; o0.z = __builtin_amdgcn_exp2f(d * d * a2[2]);
        d = xv - c[3]; o0.w = __builtin_amdgcn_exp2f(d * d * a2[3]);
        d = xv - c[4]; o1.x = __builtin_amdgcn_exp2f(d * d * a2[4]);
        d = xv - c[5]; o1.y = __builtin_amdgcn_exp2f(d * d * a2[5]);
        d = xv - c[6]; o1.z = __builtin_amdgcn_exp2f(d * d * a2[6]);
        d = xv - c[7]; o1.w = __builtin_amdgcn_exp2f(d * d * a2[7]);

        float* obase = out + (size_t)k * OUT_COLS + tid * NUM_MF;
        __builtin_nontemporal_store(o0, (v4f*)obase);        // 134 MB write-only stream:
        __builtin_nontemporal_store(o1, (v4f*)(obase + 4));  // keep it out of the caches
    }
}

extern "C" void kernel_launch(void* const* d_in, const int* in_sizes, int n_in,
                              void* d_out, int out_size, void* d_ws, size_t ws_size,
                              hipStream_t stream) {
    (void)in_sizes; (void)n_in; (void)out_size; (void)d_ws; (void)ws_size;
    const float* x       = (const float*)d_in[0];
    const float* centers = (const float*)d_in[1];
    const float* sigmas  = (const float*)d_in[2];
    float*       out     = (float*)d_out;
    fuzzify_kernel<<<GRID, BLOCK, 0, stream>>>(x, centers, sigmas, out);
}